// RandomMultiContrastRemap_58488864637435
// MI455X (gfx1250) — compile-verified
//
#include <hip/hip_runtime.h>
#include <math.h>

#define ZD 160
#define YH 192
#define XW 160
#define NN (ZD*YH*XW)
#define NBQ 4096
#define EPSF 1.1920929e-07f
#define BLK 256
#define GRD 2048
#define FLOOD_ITERS 160
#define EM_ITERS 40
#define CH 1024                 // floats per async chunk; NN == 4800*CH exactly

#if __has_builtin(__builtin_amdgcn_global_load_async_to_lds_b128) && \
    __has_builtin(__builtin_amdgcn_global_load_async_to_lds_b32)  && \
    __has_builtin(__builtin_amdgcn_s_wait_asynccnt)
#define USE_ASYNC 1
#else
#define USE_ASYNC 0
#endif

typedef __attribute__((ext_vector_type(2))) float v2f;
typedef __attribute__((ext_vector_type(8))) float v8f;

#if USE_ASYNC
// exact parameter types per hipcc diagnostics:
//   b128: (AS1 int __vector(4)*, AS3 ..., imm, imm)   b32: (AS1 int*, AS3 ..., imm, imm)
typedef int v4i_gcc __attribute__((vector_size(16)));
typedef __attribute__((address_space(1))) v4i_gcc* gas_v4p;
typedef __attribute__((address_space(1))) int*     gas_i1p;
typedef __attribute__((address_space(3))) v4i_gcc* las_v4p;
typedef __attribute__((address_space(3))) int*     las_i1p;
#endif

struct Scalars {
  unsigned umin, umax;          // encoded img min/max
  float minv, maxv;
  float qlo, qhi;
  unsigned masked_cnt;
  float n_masked;
  float q05, q95;
  float centers[3];
  float cnt[3];
  float pi[3], mu[3], var[3];
  float ppi[3], pmu[3], pvar[3];   // params used by the last EM E-step (iter 39)
  int   order[3];
  float stats[9];                  // EM accumulators: Nk[3], Sum x[3], Sum x^2[3]
  float lut[128];
  float biasnorm;
  unsigned uvmin, uvmax;
  float vmin, vmax;
};

__device__ __forceinline__ int iminc(int a,int b){return a<b?a:b;}
__device__ __forceinline__ int imaxc(int a,int b){return a>b?a:b;}
// order-preserving float<->uint encoding for atomic min/max
__device__ __forceinline__ unsigned fenc(float f){unsigned u=__float_as_uint(f); return (u&0x80000000u)?~u:(u|0x80000000u);}
__device__ __forceinline__ float fdec(unsigned e){return (e&0x80000000u)?__uint_as_float(e&0x7fffffffu):__uint_as_float(~e);}

// ---------------------------------------------------------------------------
// Block-level reduction of NS (<=16) f32 statistics using the CDNA5 matrix
// pipe: wave shuffles -> LDS tile A[16 stats][8 waves] -> two accumulating
// V_WMMA_F32_16X16X4_F32 with an all-ones B give row sums (exact f32 FMA).
// Requires blockDim.x == 256 (8 wave32s), all threads participate.
// ---------------------------------------------------------------------------
template<int NS>
__device__ void block_reduce_stats(float* acc, float* smem /*128 floats*/, float* gstats){
  const int tid = threadIdx.x;
  const int lane = tid & 31;
  const int wv = tid >> 5;
  #pragma unroll
  for (int s = 0; s < NS; ++s){
    float v = acc[s];
    v += __shfl_down(v, 16); v += __shfl_down(v, 8);
    v += __shfl_down(v, 4);  v += __shfl_down(v, 2); v += __shfl_down(v, 1);
    acc[s] = v;
  }
  if (tid < 128) smem[tid] = 0.0f;
  __syncthreads();
  if (lane == 0){
    #pragma unroll
    for (int s = 0; s < NS; ++s) smem[s*8 + wv] = acc[s];
  }
  __syncthreads();
  if (wv == 0){                       // whole wave0: EXEC all ones for WMMA
    v8f c = {};
    const int m  = lane & 15;          // A-matrix row = statistic index
    const int kb = (lane >> 4) << 1;   // lanes16-31 carry K=2,3 per ISA layout
    #pragma unroll
    for (int t = 0; t < 2; ++t){       // K blocks: waves 0..3, then 4..7
      v2f a; a.x = smem[m*8 + t*4 + kb]; a.y = smem[m*8 + t*4 + kb + 1];
      v2f bb; bb.x = 1.0f; bb.y = 1.0f;
      c = __builtin_amdgcn_wmma_f32_16x16x4_f32(false, a, false, bb,
                                                (short)0, c, false, false);
    }
    // D layout: lanes0-15 VGPR j -> M=j; lanes16-31 VGPR j -> M=j+8 (N=lane)
    if (lane == 0){
      #pragma unroll
      for (int s = 0; s < (NS < 8 ? NS : 8); ++s) atomicAdd(&gstats[s], c[s]);
    }
    if (NS > 8 && lane == 16){
      #pragma unroll
      for (int s = 8; s < NS; ++s) atomicAdd(&gstats[s], c[s-8]);
    }
  }
  __syncthreads();
}

// ---------------------------------------------------------------------------
__global__ void k_init(unsigned* hist, Scalars* sc){
  for (int b = threadIdx.x; b < NBQ; b += blockDim.x) hist[b] = 0u;
  if (threadIdx.x == 0){
    sc->umin = 0xFFFFFFFFu; sc->umax = 0u;
    sc->uvmin = 0xFFFFFFFFu; sc->uvmax = 0u;
    sc->masked_cnt = 0u;
    for (int k = 0; k < 9; ++k) sc->stats[k] = 0.0f;
  }
}

__global__ void k_minmax(const float* __restrict__ img, Scalars* sc){
  float mn = INFINITY, mx = -INFINITY;
  const int stride = gridDim.x * blockDim.x;
  for (int i = blockIdx.x*blockDim.x + threadIdx.x; i < NN; i += stride){
    float v = img[i]; mn = fminf(mn, v); mx = fmaxf(mx, v);
  }
  for (int o = 16; o; o >>= 1){ mn = fminf(mn, __shfl_down(mn,o)); mx = fmaxf(mx, __shfl_down(mx,o)); }
  if ((threadIdx.x & 31) == 0){ atomicMin(&sc->umin, fenc(mn)); atomicMax(&sc->umax, fenc(mx)); }
}

__global__ void k_range(Scalars* sc){
  float mn = fdec(sc->umin), mx = fdec(sc->umax);
  if (!(mx > mn)) mx = mn + 1e-6f;
  sc->minv = mn; sc->maxv = mx;
}

__global__ void k_hist_img(const float* __restrict__ img, unsigned* hist, const Scalars* sc){
  __shared__ unsigned sh[NBQ];
  for (int b = threadIdx.x; b < NBQ; b += blockDim.x) sh[b] = 0u;
  __syncthreads();
  const float mn = sc->minv;
  const float scale = (float)NBQ / (sc->maxv - mn);
  const int stride = gridDim.x * blockDim.x;
  for (int i = blockIdx.x*blockDim.x + threadIdx.x; i < NN; i += stride){
    int b = (int)((img[i] - mn) * scale);
    b = iminc(imaxc(b, 0), NBQ-1);
    atomicAdd(&sh[b], 1u);
  }
  __syncthreads();
  for (int b = threadIdx.x; b < NBQ; b += blockDim.x) if (sh[b]) atomicAdd(&hist[b], sh[b]);
}

__device__ float hist_q(const unsigned* h, float target, float lo, float binw){
  float c = 0.0f;
  for (int b = 0; b < NBQ; ++b){
    float hb = (float)h[b];
    if (c + hb > target){
      float fr = (target - c) / fmaxf(hb, 1.0f);
      return lo + ((float)b + fr) * binw;
    }
    c += hb;
  }
  return lo + (float)NBQ * binw;
}

__global__ void k_q_img(const unsigned* hist, Scalars* sc){
  const float binw = (sc->maxv - sc->minv) / (float)NBQ;
  sc->qlo = hist_q(hist, 0.01f * (float)(NN-1), sc->minv, binw);
  sc->qhi = hist_q(hist, 0.99f * (float)(NN-1), sc->minv, binw);
}

__global__ void k_normalize(const float* __restrict__ img, const Scalars* sc,
                            float* __restrict__ vol, unsigned char* __restrict__ m0){
  const float qlo = sc->qlo;
  const float inv = 1.0f / fmaxf(sc->qhi - sc->qlo, EPSF);
  const int stride = gridDim.x * blockDim.x;
  for (int i = blockIdx.x*blockDim.x + threadIdx.x; i < NN; i += stride){
    float x = fminf(fmaxf((img[i] - qlo) * inv, 0.0f), 1.0f);
    vol[i] = x;
    m0[i] = (x > 0.0f) ? 1 : 0;
  }
}

__global__ void k_dilate(const unsigned char* __restrict__ in, unsigned char* __restrict__ out){
  const int stride = gridDim.x * blockDim.x;
  for (int i = blockIdx.x*blockDim.x + threadIdx.x; i < NN; i += stride){
    int z = i / (YH*XW); int r = i - z*YH*XW; int y = r / XW; int x = r - y*XW;
    unsigned char m = 0;
    for (int dz = -1; dz <= 1; ++dz){ int zz = z+dz; if ((unsigned)zz >= (unsigned)ZD) continue;
      for (int dy = -1; dy <= 1; ++dy){ int yy = y+dy; if ((unsigned)yy >= (unsigned)YH) continue;
        const unsigned char* row = &in[(zz*YH + yy)*XW];
        for (int dx = -1; dx <= 1; ++dx){ int xx = x+dx; if ((unsigned)xx >= (unsigned)XW) continue;
          m |= row[xx];
        }}}
    out[i] = m ? 1 : 0;
  }
}

__global__ void k_erode(const unsigned char* __restrict__ in, unsigned char* __restrict__ out){
  const int stride = gridDim.x * blockDim.x;
  for (int i = blockIdx.x*blockDim.x + threadIdx.x; i < NN; i += stride){
    int z = i / (YH*XW); int r = i - z*YH*XW; int y = r / XW; int x = r - y*XW;
    unsigned char a = 1;
    for (int dz = -1; dz <= 1; ++dz){ int zz = z+dz; if ((unsigned)zz >= (unsigned)ZD) continue;
      for (int dy = -1; dy <= 1; ++dy){ int yy = y+dy; if ((unsigned)yy >= (unsigned)YH) continue;
        const unsigned char* row = &in[(zz*YH + yy)*XW];
        for (int dx = -1; dx <= 1; ++dx){ int xx = x+dx; if ((unsigned)xx >= (unsigned)XW) continue;
          a &= row[xx];
        }}}
    out[i] = a ? 1 : 0;
  }
}

__global__ void k_flood_init(const unsigned char* __restrict__ closed, unsigned char* __restrict__ out){
  const int stride = gridDim.x * blockDim.x;
  for (int i = blockIdx.x*blockDim.x + threadIdx.x; i < NN; i += stride){
    int z = i / (YH*XW); int r = i - z*YH*XW; int y = r / XW; int x = r - y*XW;
    int border = (z==0)|(z==ZD-1)|(y==0)|(y==YH-1)|(x==0)|(x==XW-1);
    out[i] = (border && !closed[i]) ? 1 : 0;
  }
}

__global__ void k_flood_step(const unsigned char* __restrict__ closed,
                             const unsigned char* __restrict__ cur,
                             unsigned char* __restrict__ nxt){
  const int stride = gridDim.x * blockDim.x;
  for (int i = blockIdx.x*blockDim.x + threadIdx.x; i < NN; i += stride){
    if (closed[i]){ nxt[i] = 0; continue; }
    int z = i / (YH*XW); int r = i - z*YH*XW; int y = r / XW; int x = r - y*XW;
    unsigned char m = 0;
    for (int dz = -1; dz <= 1; ++dz){ int zz = z+dz; if ((unsigned)zz >= (unsigned)ZD) continue;
      for (int dy = -1; dy <= 1; ++dy){ int yy = y+dy; if ((unsigned)yy >= (unsigned)YH) continue;
        const unsigned char* row = &cur[(zz*YH + yy)*XW];
        for (int dx = -1; dx <= 1; ++dx){ int xx = x+dx; if ((unsigned)xx >= (unsigned)XW) continue;
          m |= row[xx];
        }}}
    nxt[i] = m ? 1 : 0;
  }
}

__global__ void k_zero_hist(unsigned* hist, Scalars* sc){
  for (int b = threadIdx.x; b < NBQ; b += blockDim.x) hist[b] = 0u;
  if (threadIdx.x == 0) sc->masked_cnt = 0u;
}

__global__ void k_mask_final(const unsigned char* __restrict__ closed,
                             const unsigned char* __restrict__ reach,
                             unsigned char* __restrict__ mask, Scalars* sc){
  int c = 0;
  const int stride = gridDim.x * blockDim.x;
  for (int i = blockIdx.x*blockDim.x + threadIdx.x; i < NN; i += stride){
    unsigned char m = (closed[i] || !reach[i]) ? 1 : 0;   // closed | holes
    mask[i] = m; c += m;
  }
  for (int o = 16; o; o >>= 1) c += __shfl_down(c, o);
  if ((threadIdx.x & 31) == 0) atomicAdd(&sc->masked_cnt, (unsigned)c);
}

__global__ void k_hist_masked(const float* __restrict__ vol, const unsigned char* __restrict__ mask,
                              unsigned* hist){
  __shared__ unsigned sh[NBQ];
  for (int b = threadIdx.x; b < NBQ; b += blockDim.x) sh[b] = 0u;
  __syncthreads();
  const int stride = gridDim.x * blockDim.x;
  for (int i = blockIdx.x*blockDim.x + threadIdx.x; i < NN; i += stride){
    if (!mask[i]) continue;
    int b = (int)(vol[i] * (float)NBQ);
    b = iminc(imaxc(b, 0), NBQ-1);
    atomicAdd(&sh[b], 1u);
  }
  __syncthreads();
  for (int b = threadIdx.x; b < NBQ; b += blockDim.x) if (sh[b]) atomicAdd(&hist[b], sh[b]);
}

__global__ void k_q_masked(const unsigned* hist, Scalars* sc){
  const float total = (float)sc->masked_cnt;
  sc->n_masked = fmaxf(total, 1.0f);
  const float binw = 1.0f / (float)NBQ;
  float q05 = hist_q(hist, 0.05f * (total - 1.0f), 0.0f, binw);
  float q95 = hist_q(hist, 0.95f * (total - 1.0f), 0.0f, binw);
  sc->q05 = q05; sc->q95 = q95;
  sc->centers[0] = q05; sc->centers[1] = 0.5f*(q05+q95); sc->centers[2] = q95;
}

__global__ void k_init_assign(const float* __restrict__ vol, const unsigned char* __restrict__ mask,
                              Scalars* sc){
  __shared__ float smem[128];
  float acc[6] = {0,0,0,0,0,0};
  const float c0 = sc->centers[0], c1 = sc->centers[1], c2 = sc->centers[2];
  const int stride = gridDim.x * blockDim.x;
  for (int i = blockIdx.x*blockDim.x + threadIdx.x; i < NN; i += stride){
    if (!mask[i]) continue;
    float x = vol[i];
    float d0 = fabsf(x-c0), d1 = fabsf(x-c1), d2 = fabsf(x-c2);
    int k = 0; float bd = d0;
    if (d1 < bd){ k = 1; bd = d1; }
    if (d2 < bd){ k = 2; }
    acc[k] += 1.0f; acc[3+k] += x;
  }
  block_reduce_stats<6>(acc, smem, sc->stats);
}

__global__ void k_init_mu(Scalars* sc){
  for (int k = 0; k < 3; ++k){
    float cnt = fmaxf(sc->stats[k], 1.0f);
    sc->cnt[k] = cnt;
    sc->pi[k]  = cnt / sc->n_masked;
    sc->mu[k]  = sc->stats[3+k] / cnt;
  }
  for (int k = 0; k < 9; ++k) sc->stats[k] = 0.0f;
}

__global__ void k_init_var(const float* __restrict__ vol, const unsigned char* __restrict__ mask,
                           Scalars* sc){
  __shared__ float smem[128];
  float acc[3] = {0,0,0};
  const float c0 = sc->centers[0], c1 = sc->centers[1], c2 = sc->centers[2];
  const float m0 = sc->mu[0], m1 = sc->mu[1], m2 = sc->mu[2];
  const int stride = gridDim.x * blockDim.x;
  for (int i = blockIdx.x*blockDim.x + threadIdx.x; i < NN; i += stride){
    if (!mask[i]) continue;
    float x = vol[i];
    float d0 = fabsf(x-c0), d1 = fabsf(x-c1), d2 = fabsf(x-c2);
    int k = 0; float bd = d0;
    if (d1 < bd){ k = 1; bd = d1; }
    if (d2 < bd){ k = 2; }
    float mu = (k==0) ? m0 : (k==1) ? m1 : m2;
    float d = x - mu;
    acc[k] += d*d;
  }
  block_reduce_stats<3>(acc, smem, sc->stats);
}

__global__ void k_init_var_fin(Scalars* sc){
  for (int k = 0; k < 3; ++k)
    sc->var[k] = fmaxf(sc->stats[k] / sc->cnt[k], 1e-4f);
  for (int k = 0; k < 9; ++k) sc->stats[k] = 0.0f;
}

// ---------------------------------------------------------------------------
// Hot loop (40 launches): EM E-step + sufficient statistics.
// Double-buffered GLOBAL_LOAD_ASYNC_TO_LDS staging: each lane async-loads
// exactly the 16B of vol (B128) + 4B of mask (B32) it will consume, so the
// per-wave S_WAIT_ASYNCCNT is the only sync needed (no barriers); the next
// 4KB chunk stays in flight while the current one is processed.
// ---------------------------------------------------------------------------
__global__ void k_em_pass(const float* __restrict__ vol, const unsigned char* __restrict__ mask,
                          Scalars* sc){
  __shared__ float smem[128];
  float acc[9] = {0,0,0,0,0,0,0,0,0};
  float mu[3], iv[3], nm[3];
  #pragma unroll
  for (int k = 0; k < 3; ++k){
    mu[k] = sc->mu[k];
    float v = sc->var[k];
    iv[k] = -0.5f / v;
    nm[k] = sc->pi[k] * rsqrtf(6.283185307179586f * v);
  }
#if USE_ASYNC
  __shared__ float bufv[2][CH];
  __shared__ unsigned char bufm[2][CH];
  const int tid = threadIdx.x;
  const int e0 = tid * 4;
  const int nchunks = NN / CH;
  int c = blockIdx.x;
  int pb = 0;
  if (c < nchunks){
    __builtin_amdgcn_global_load_async_to_lds_b128(
      (gas_v4p)(vol + c*CH + e0), (las_v4p)(&bufv[0][e0]), 0, 0);
    __builtin_amdgcn_global_load_async_to_lds_b32(
      (gas_i1p)(mask + c*CH + e0), (las_i1p)(&bufm[0][e0]), 0, 0);
  }
  for (; c < nchunks; c += gridDim.x){
    const int cn = c + gridDim.x;
    if (cn < nchunks){
      __builtin_amdgcn_global_load_async_to_lds_b128(
        (gas_v4p)(vol + cn*CH + e0), (las_v4p)(&bufv[pb^1][e0]), 0, 0);
      __builtin_amdgcn_global_load_async_to_lds_b32(
        (gas_i1p)(mask + cn*CH + e0), (las_i1p)(&bufm[pb^1][e0]), 0, 0);
      __builtin_amdgcn_s_wait_asynccnt(2);   // current chunk done, next in flight
    } else {
      __builtin_amdgcn_s_wait_asynccnt(0);
    }
    #pragma unroll
    for (int j = 0; j < 4; ++j){
      if (!bufm[pb][e0 + j]) continue;
      float x = bufv[pb][e0 + j];
      float d0 = x-mu[0], d1 = x-mu[1], d2 = x-mu[2];
      float p0 = fmaxf(nm[0]*__expf(iv[0]*d0*d0), 1e-12f);
      float p1 = fmaxf(nm[1]*__expf(iv[1]*d1*d1), 1e-12f);
      float p2 = fmaxf(nm[2]*__expf(iv[2]*d2*d2), 1e-12f);
      float inv_s = 1.0f / (p0 + p1 + p2);
      float q0 = p0*inv_s, q1 = p1*inv_s, q2 = p2*inv_s;
      float xx = x*x;
      acc[0] += q0;    acc[1] += q1;    acc[2] += q2;
      acc[3] += q0*x;  acc[4] += q1*x;  acc[5] += q2*x;
      acc[6] += q0*xx; acc[7] += q1*xx; acc[8] += q2*xx;
    }
    pb ^= 1;
  }
#else
  const int stride = gridDim.x * blockDim.x;
  for (int i = blockIdx.x*blockDim.x + threadIdx.x; i < NN; i += stride){
    __builtin_prefetch(&vol[i + stride], 0, 0);     // global_prefetch_b8
    if (!mask[i]) continue;
    float x = vol[i];
    float d0 = x-mu[0], d1 = x-mu[1], d2 = x-mu[2];
    float p0 = fmaxf(nm[0]*__expf(iv[0]*d0*d0), 1e-12f);
    float p1 = fmaxf(nm[1]*__expf(iv[1]*d1*d1), 1e-12f);
    float p2 = fmaxf(nm[2]*__expf(iv[2]*d2*d2), 1e-12f);
    float inv_s = 1.0f / (p0 + p1 + p2);
    float q0 = p0*inv_s, q1 = p1*inv_s, q2 = p2*inv_s;
    float xx = x*x;
    acc[0] += q0;    acc[1] += q1;    acc[2] += q2;
    acc[3] += q0*x;  acc[4] += q1*x;  acc[5] += q2*x;
    acc[6] += q0*xx; acc[7] += q1*xx; acc[8] += q2*xx;
  }
#endif
  block_reduce_stats<9>(acc, smem, sc->stats);
}

__global__ void k_em_fin(Scalars* sc, int last){
  float S0[3], S1[3], S2[3];
  for (int k = 0; k < 3; ++k){ S0[k]=sc->stats[k]; S1[k]=sc->stats[3+k]; S2[k]=sc->stats[6+k]; }
  for (int k = 0; k < 3; ++k){ sc->ppi[k]=sc->pi[k]; sc->pmu[k]=sc->mu[k]; sc->pvar[k]=sc->var[k]; }
  for (int k = 0; k < 3; ++k){
    float Nk = fmaxf(S0[k], 1e-12f);
    float mu = S1[k] / Nk;
    float vr = (S2[k] - 2.0f*mu*S1[k] + mu*mu*S0[k]) / Nk;
    sc->pi[k]  = Nk / sc->n_masked;
    sc->mu[k]  = mu;
    sc->var[k] = fmaxf(vr, 1e-4f);
  }
  for (int k = 0; k < 9; ++k) sc->stats[k] = 0.0f;
  if (last){
    int o0=0,o1=1,o2=2, t;
    if (sc->mu[o0] > sc->mu[o1]){ t=o0; o0=o1; o1=t; }
    if (sc->mu[o1] > sc->mu[o2]){ t=o1; o1=o2; o2=t; }
    if (sc->mu[o0] > sc->mu[o1]){ t=o0; o0=o1; o1=t; }
    sc->order[0]=o0; sc->order[1]=o1; sc->order[2]=o2;
  }
}

__global__ void k_remap(float* __restrict__ vol, const unsigned char* __restrict__ mask,
                        const float* __restrict__ targets, const Scalars* sc){
  float mu[3], iv[3], nm[3], tg[3]; int ord[3];
  #pragma unroll
  for (int k = 0; k < 3; ++k){
    mu[k] = sc->pmu[k];
    float v = sc->pvar[k];
    iv[k] = -0.5f / v;
    nm[k] = sc->ppi[k] * rsqrtf(6.283185307179586f * v);
    ord[k] = sc->order[k];
    tg[k] = targets[k];
  }
  const int stride = gridDim.x * blockDim.x;
  for (int i = blockIdx.x*blockDim.x + threadIdx.x; i < NN; i += stride){
    if (!mask[i]) continue;
    float x = vol[i];
    float p[3];
    float s = 0.0f;
    #pragma unroll
    for (int k = 0; k < 3; ++k){
      float d = x - mu[k];
      p[k] = fmaxf(nm[k]*__expf(iv[k]*d*d), 1e-12f);
      s += p[k];
    }
    float f[3]; float fs = 0.0f;
    #pragma unroll
    for (int k = 0; k < 3; ++k){ f[k] = sqrtf(p[k]/s); fs += f[k]; }
    fs = fmaxf(fs, 1e-12f);
    float remap = (tg[0]*f[ord[0]] + tg[1]*f[ord[1]] + tg[2]*f[ord[2]]) / fs;
    vol[i] = 0.9f*remap + 0.1f*x;
  }
}

__global__ void k_lut_build(const float* __restrict__ raw, Scalars* sc){
  const int K = 77, HK = 38;           // k = (int)(6*12.8)+1 | 1
  const float sig = 12.8f;             // NUM_BINS/10
  const int i = threadIdx.x;           // 0..127
  float gsum = 0.0f;
  for (int j = 0; j < K; ++j){ float xs = (float)(j-HK)/sig; gsum += __expf(-0.5f*xs*xs); }
  float accv = 0.0f;
  for (int j = 0; j < K; ++j){
    float xs = (float)(j-HK)/sig;
    float g = __expf(-0.5f*xs*xs) / gsum;
    int src = i + j - HK;
    if (src >= 0 && src < 128) accv += g * raw[src];
  }
  __shared__ float t[128];
  __shared__ float mn, mx;
  t[i] = accv;
  __syncthreads();
  if (i == 0){
    float a = t[0], b = t[0];
    for (int j = 1; j < 128; ++j){ a = fminf(a, t[j]); b = fmaxf(b, t[j]); }
    mn = a; mx = b;
  }
  __syncthreads();
  float v = fminf(fmaxf((t[i]-mn)/fmaxf(mx-mn, EPSF), 0.0f), 1.0f);
  if (i == 0)   v = 0.0f;
  if (i == 127) v = 1.0f;
  sc->lut[i] = v;
}

__global__ void k_biasnorm(const float* __restrict__ cf, Scalars* sc){
  const float c0 = cf[0], c1 = cf[1], c2 = cf[2];
  int dims[3] = { ZD, YH, XW };
  float sumMax = 0.0f, sumMin = 0.0f;
  for (int a = 0; a < 3; ++a){
    float mx = -INFINITY, mn = INFINITY;
    for (int i = 0; i < dims[a]; ++i){
      float t = -1.0f + 2.0f*(float)i/(float)(dims[a]-1);
      float g = t*(c0 + t*(c1 + t*c2));
      mx = fmaxf(mx, g); mn = fminf(mn, g);
    }
    sumMax += mx; sumMin += mn;
  }
  // separable field: max|field| = max(maxsum, -minsum)
  sc->biasnorm = fmaxf(fmaxf(sumMax, -sumMin), 1e-6f);
}

__global__ void k_apply(const float* __restrict__ vol, const float* __restrict__ noise,
                        const float* __restrict__ cf, const unsigned char* __restrict__ mask,
                        float* __restrict__ out, Scalars* sc){
  __shared__ float slut[128];
  if (threadIdx.x < 128) slut[threadIdx.x] = sc->lut[threadIdx.x];
  __syncthreads();
  const float c0 = cf[0], c1 = cf[1], c2 = cf[2];
  const float bscale = 0.3f / sc->biasnorm;
  float mn = INFINITY, mx = -INFINITY;
  const int stride = gridDim.x * blockDim.x;
  for (int i = blockIdx.x*blockDim.x + threadIdx.x; i < NN; i += stride){
    int z = i / (YH*XW); int r = i - z*YH*XW; int y = r / XW; int x = r - y*XW;
    float tz = -1.0f + 2.0f*(float)z/(float)(ZD-1);
    float ty = -1.0f + 2.0f*(float)y/(float)(YH-1);
    float tx = -1.0f + 2.0f*(float)x/(float)(XW-1);
    float field = tz*(c0 + tz*(c1 + tz*c2)) + ty*(c0 + ty*(c1 + ty*c2)) + tx*(c0 + tx*(c1 + tx*c2));
    float bias = 1.0f + bscale*field;
    float v = vol[i];
    float idx = v * 127.0f;
    int lo = (int)floorf(idx);
    lo = iminc(imaxc(lo, 0), 126);
    float wf = idx - (float)lo;
    float lv = slut[lo]*(1.0f-wf) + slut[lo+1]*wf;
    float val = lv*bias + 0.03f*noise[i];
    out[i] = val;
    if (mask[i]){ mn = fminf(mn, val); mx = fmaxf(mx, val); }
  }
  for (int o = 16; o; o >>= 1){ mn = fminf(mn, __shfl_down(mn,o)); mx = fmaxf(mx, __shfl_down(mx,o)); }
  if ((threadIdx.x & 31) == 0){ atomicMin(&sc->uvmin, fenc(mn)); atomicMax(&sc->uvmax, fenc(mx)); }
}

__global__ void k_vfin(Scalars* sc){
  sc->vmin = fdec(sc->uvmin);
  sc->vmax = fdec(sc->uvmax);
}

__global__ void k_final(float* __restrict__ out, const unsigned char* __restrict__ mask,
                        const Scalars* sc){
  const float vmin = sc->vmin;
  const float inv = 1.0f / fmaxf(sc->vmax - sc->vmin, EPSF);
  const int stride = gridDim.x * blockDim.x;
  for (int i = blockIdx.x*blockDim.x + threadIdx.x; i < NN; i += stride){
    float v = fminf(fmaxf((out[i]-vmin)*inv, 0.0f), 1.0f);
    out[i] = mask[i] ? v : 0.0f;
  }
}

// ---------------------------------------------------------------------------
extern "C" void kernel_launch(void* const* d_in, const int* in_sizes, int n_in,
                              void* d_out, int out_size, void* d_ws, size_t ws_size,
                              hipStream_t stream){
  (void)in_sizes; (void)n_in; (void)out_size; (void)ws_size;
  const float* img      = (const float*)d_in[0];   // [1,1,D,H,W]
  const float* lut_raw  = (const float*)d_in[1];   // [128]
  const float* targets  = (const float*)d_in[2];   // [3] sorted
  const float* coeffs   = (const float*)d_in[3];   // [3]
  const float* noise    = (const float*)d_in[4];   // [1,1,D,H,W]
  float* out            = (float*)d_out;           // [1,1,D,H,W]

  char* ws = (char*)d_ws;
  float* vol = (float*)ws;                                        // N f32 (19.7MB)
  unsigned char* mA = (unsigned char*)(ws + (size_t)NN*sizeof(float)); // dil -> final mask
  unsigned char* mB = mA + NN;                                    // closed
  unsigned char* mC = mB + NN;                                    // flood ping
  unsigned char* mD = mC + NN;                                    // flood pong
  unsigned* hist = (unsigned*)(mD + NN);                          // NBQ bins
  Scalars* sc = (Scalars*)(hist + NBQ);

  dim3 g(GRD), b(BLK);
  // 1) robust quantile normalization
  k_init<<<1,256,0,stream>>>(hist, sc);
  k_minmax<<<g,b,0,stream>>>(img, sc);
  k_range<<<1,1,0,stream>>>(sc);
  k_hist_img<<<g,b,0,stream>>>(img, hist, sc);
  k_q_img<<<1,1,0,stream>>>(hist, sc);
  k_normalize<<<g,b,0,stream>>>(img, sc, vol, mC);
  // 2) morphological closing + border-connected flood fill
  k_dilate<<<g,b,0,stream>>>(mC, mA);
  k_erode<<<g,b,0,stream>>>(mA, mB);
  k_flood_init<<<g,b,0,stream>>>(mB, mC);
  unsigned char* cur = mC; unsigned char* nxt = mD;
  for (int it = 0; it < FLOOD_ITERS; ++it){
    k_flood_step<<<g,b,0,stream>>>(mB, cur, nxt);
    unsigned char* t = cur; cur = nxt; nxt = t;
  }
  k_zero_hist<<<1,256,0,stream>>>(hist, sc);
  k_mask_final<<<g,b,0,stream>>>(mB, cur, mA, sc);
  // 3) GMM: init from masked 5/95 quantiles, then EM with WMMA-reduced stats
  k_hist_masked<<<g,b,0,stream>>>(vol, mA, hist);
  k_q_masked<<<1,1,0,stream>>>(hist, sc);
  k_init_assign<<<g,b,0,stream>>>(vol, mA, sc);
  k_init_mu<<<1,1,0,stream>>>(sc);
  k_init_var<<<g,b,0,stream>>>(vol, mA, sc);
  k_init_var_fin<<<1,1,0,stream>>>(sc);
  for (int it = 0; it < EM_ITERS; ++it){
    k_em_pass<<<g,b,0,stream>>>(vol, mA, sc);
    k_em_fin<<<1,1,0,stream>>>(sc, (it == EM_ITERS-1) ? 1 : 0);
  }
  k_remap<<<g,b,0,stream>>>(vol, mA, targets, sc);
  // 4-5) LUT contrast + polynomial bias + noise
  k_lut_build<<<1,128,0,stream>>>(lut_raw, sc);
  k_biasnorm<<<1,1,0,stream>>>(coeffs, sc);
  k_apply<<<g,b,0,stream>>>(vol, noise, coeffs, mA, out, sc);
  // 6) masked min/max renormalization
  k_vfin<<<1,1,0,stream>>>(sc);
  k_final<<<g,b,0,stream>>>(out, mA, sc);
}